// MSE_topK_loss_88201448391196
// MI455X (gfx1250) — compile-verified
//
#include <hip/hip_runtime.h>

// Problem geometry (B=2, C=4, D=H=W=128)
#define ROWS        8u
#define ROW_N       (1u << 21)                 // 2,097,152 voxels per row
#define TOTAL_ELEMS (ROWS * ROW_N)             // 16,777,216
#define TOPN        209715u                    // round(ROW_N * 0.10)
#define HBINS       4096
#define CHUNK       4096u                      // elements per TDM tile (16 KB)
#define CPB         8u                         // chunks per block (divides 512)

typedef unsigned int uint32x4 __attribute__((ext_vector_type(4)));
typedef int          int32x4  __attribute__((ext_vector_type(4)));
typedef int          int32x8  __attribute__((ext_vector_type(8)));

__device__ __forceinline__ float sigmoid_f(float v) {
  return 1.0f / (1.0f + __expf(-v));
}

__device__ __forceinline__ float sq_err(float xv, float tv) {
  float d = sigmoid_f(xv) - tv;
  return d * d;                                // in [0,1) -> bits <= 0x3F800000
}

// ---------------------------------------------------------------------------
// Pass 1/2/3: per-row histogram of a bit field of the f32 bit pattern.
// LDS-privatized (ds_add_u32), b128 global loads + global_prefetch_b8,
// global atomic flush.
// ---------------------------------------------------------------------------
__global__ void __launch_bounds__(256) hist_kernel(
    const float* __restrict__ x, const float* __restrict__ t,
    unsigned* __restrict__ hist, const unsigned* __restrict__ prefix,
    unsigned match_mask, unsigned bin_shift, unsigned bin_mask) {
  __shared__ unsigned lh[HBINS];
  const unsigned tid = threadIdx.x;
  for (unsigned b = tid; b < HBINS; b += 256u) lh[b] = 0u;
  __syncthreads();

  const unsigned row  = blockIdx.x >> 8;       // 256 blocks per row (8192 elems/block)
  const unsigned pfx  = prefix[row] & match_mask;
  const float4* __restrict__ x4 = reinterpret_cast<const float4*>(x);
  const float4* __restrict__ t4 = reinterpret_cast<const float4*>(t);
  const unsigned base4 = blockIdx.x * 2048u;   // float4 units

#pragma unroll
  for (int i = 0; i < 8; ++i) {
    const unsigned idx = base4 + (unsigned)i * 256u + tid;
    if (i < 6) {                               // stage next iterations into cache
      __builtin_prefetch(&x4[idx + 512u], 0, 1);
      __builtin_prefetch(&t4[idx + 512u], 0, 1);
    }
    const float4 xv = x4[idx];
    const float4 tv = t4[idx];
    const float v[4] = { sq_err(xv.x, tv.x), sq_err(xv.y, tv.y),
                         sq_err(xv.z, tv.z), sq_err(xv.w, tv.w) };
#pragma unroll
    for (int j = 0; j < 4; ++j) {
      const unsigned bits = __float_as_uint(v[j]);
      if ((bits & match_mask) == pfx)
        atomicAdd(&lh[(bits >> bin_shift) & bin_mask], 1u);
    }
  }
  __syncthreads();

  unsigned* hrow = hist + row * HBINS;
  for (unsigned b = tid; b < HBINS; b += 256u) {
    const unsigned c = lh[b];
    if (c) atomicAdd(&hrow[b], c);
  }
}

// ---------------------------------------------------------------------------
// Radix-select step: one 256-thread block per row. Parallel suffix scan from
// the top bin finds the bin containing the k-th largest value; updates the
// bit prefix and the residual count (exact tie handling).
// ---------------------------------------------------------------------------
__global__ void __launch_bounds__(256) select_kernel(
    const unsigned* __restrict__ hist, unsigned* __restrict__ prefix,
    unsigned* __restrict__ remaining, unsigned bin_shift, int level, int nbins) {
  __shared__ unsigned ssum[256];
  const unsigned row = blockIdx.x;
  const unsigned tid = threadIdx.x;
  const int bpt = nbins >> 8;                  // bins per thread: 16 (L0/L1) or 1 (L2)
  const unsigned* h = hist + row * HBINS;

  // Read residual target ONCE before any thread can update it.
  const unsigned rem = (level == 0) ? TOPN : remaining[row];

  unsigned s = 0;
  for (int j = 0; j < bpt; ++j) s += h[tid * bpt + j];
  ssum[tid] = s;
  __syncthreads();

  unsigned suffix_incl = 0;                    // counts in bins >= my chunk start
  for (int k = (int)tid; k < 256; ++k) suffix_incl += ssum[k];
  const unsigned suffix_excl = suffix_incl - s; // counts strictly above my chunk

  if (rem > suffix_excl && rem <= suffix_incl) { // exactly one thread wins
    unsigned cum = suffix_excl;
    int sel = (int)tid * bpt;
    for (int b = (int)tid * bpt + bpt - 1; b >= (int)tid * bpt; --b) {
      const unsigned c = h[b];
      if (cum + c >= rem) { sel = b; break; }
      cum += c;
    }
    prefix[row]    |= ((unsigned)sel << bin_shift);
    remaining[row]  = rem - cum;               // copies of threshold value to keep
  }
}

// ---------------------------------------------------------------------------
// Tensor Data Mover: DMA a contiguous run of n_elems f32 into LDS.
// D# built per CDNA5 ISA (group0 + group1; groups 2/3 zero).
// ---------------------------------------------------------------------------
__device__ __forceinline__ void tdm_load_1d_f32(unsigned long long gaddr,
                                                unsigned lds_byte_off,
                                                unsigned n_elems) {
#if defined(__HIP_DEVICE_COMPILE__)
  const unsigned lo = (unsigned)__builtin_amdgcn_readfirstlane((int)(unsigned)(gaddr & 0xFFFFFFFFull));
  const unsigned hi = (unsigned)__builtin_amdgcn_readfirstlane((int)(unsigned)(gaddr >> 32));
  uint32x4 g0;
  g0[0] = 1u;                                     // count=1 (valid), user mode
  g0[1] = lds_byte_off;                           // lds_addr
  g0[2] = lo;                                     // global_addr[31:0]
  g0[3] = (hi & 0x01FFFFFFu) | 0x80000000u;       // global_addr[56:32] | type=2
  int32x8 g1;
  g1[0] = 0x00020000;                             // data_size=2 (4 bytes)
  g1[1] = (int)((n_elems & 0xFFFFu) << 16);       // tensor_dim0[15:0]
  g1[2] = (int)(((n_elems >> 16) & 0xFFFFu) | (1u << 16)); // dim0[31:16], tensor_dim1=1
  g1[3] = (int)((n_elems & 0xFFFFu) << 16);       // tile_dim0
  g1[4] = 1;                                      // tile_dim1 = 1
  g1[5] = (int)n_elems;                           // tensor_dim0_stride[31:0]
  g1[6] = 0;
  g1[7] = 0;
  const int32x4 gz = {0, 0, 0, 0};
#if __clang_major__ >= 23
  const int32x8 gz8 = {0, 0, 0, 0, 0, 0, 0, 0};
  __builtin_amdgcn_tensor_load_to_lds(g0, g1, gz, gz, gz8, 0);
#else
  __builtin_amdgcn_tensor_load_to_lds(g0, g1, gz, gz, 0);
#endif
#else
  (void)gaddr; (void)lds_byte_off; (void)n_elems;
#endif
}

// ---------------------------------------------------------------------------
// Final streaming pass, software-pipelined TDM double-buffer:
//   issue pair(j+1) -> s_wait_tensorcnt(2) (pair j done, in-order per wave)
//   -> consume buffer j&1. Single-wave blocks own their TENSORcnt; wave32
//   shfl reduction + one f64 global atomic per block.
// LDS layout (dynamic, 64 KB): buf0 {x,t} at 0/16KB, buf1 {x,t} at 32/48KB.
// ---------------------------------------------------------------------------
__global__ void __launch_bounds__(32) topk_sum_kernel(
    const float* __restrict__ x, const float* __restrict__ t,
    const unsigned* __restrict__ prefix, double* __restrict__ sums) {
  extern __shared__ float smem[];
  const unsigned chunk0 = blockIdx.x * CPB;    // CPB=8 divides 512 -> one row/block
  const unsigned row    = chunk0 >> 9;         // 512 chunks per row
  const unsigned lane   = threadIdx.x;
  const unsigned thr    = prefix[row];         // exact f32 bit-pattern threshold

  const unsigned long long xb = (unsigned long long)(uintptr_t)x;
  const unsigned long long tb = (unsigned long long)(uintptr_t)t;
  const unsigned CB = CHUNK * (unsigned)sizeof(float);   // 16384 bytes per tile

  // Prologue: stage chunk 0 into buffer 0.
  tdm_load_1d_f32(xb + (unsigned long long)chunk0 * CB, 0u, CHUNK);
  tdm_load_1d_f32(tb + (unsigned long long)chunk0 * CB, CB, CHUNK);

  float acc = 0.0f;
#pragma unroll
  for (unsigned j = 0; j < CPB; ++j) {
    if (j + 1u < CPB) {
      const unsigned buf = (j + 1u) & 1u;      // issue next pair into other buffer
      const unsigned long long off = (unsigned long long)(chunk0 + j + 1u) * CB;
      tdm_load_1d_f32(xb + off, buf * 2u * CB, CHUNK);
      tdm_load_1d_f32(tb + off, buf * 2u * CB + CB, CHUNK);
#if defined(__HIP_DEVICE_COMPILE__)
      __builtin_amdgcn_s_wait_tensorcnt(2);    // <=2 outstanding => pair j complete
#endif
    } else {
#if defined(__HIP_DEVICE_COMPILE__)
      __builtin_amdgcn_s_wait_tensorcnt(0);    // drain for the final chunk
#endif
    }

    const float* bx = smem + (size_t)(j & 1u) * 2u * CHUNK;
    const float* bt = bx + CHUNK;
    const float4* sx4 = reinterpret_cast<const float4*>(bx);
    const float4* st4 = reinterpret_cast<const float4*>(bt);
#pragma unroll 4
    for (unsigned i = lane; i < CHUNK / 4u; i += 32u) {
      const float4 xv = sx4[i];
      const float4 tv = st4[i];
      const float v0 = sq_err(xv.x, tv.x);
      const float v1 = sq_err(xv.y, tv.y);
      const float v2 = sq_err(xv.z, tv.z);
      const float v3 = sq_err(xv.w, tv.w);
      if (__float_as_uint(v0) > thr) acc += v0;
      if (__float_as_uint(v1) > thr) acc += v1;
      if (__float_as_uint(v2) > thr) acc += v2;
      if (__float_as_uint(v3) > thr) acc += v3;
    }
  }

  for (int off = 16; off > 0; off >>= 1)
    acc += __shfl_xor(acc, off, 32);
  if (lane == 0)
    atomicAdd(&sums[row], (double)acc);
}

// ---------------------------------------------------------------------------
// Combine: mean = (Σ_r sum_r + remaining_r * thr_value_r) / (ROWS * TOPN)
// ---------------------------------------------------------------------------
__global__ void finalize_kernel(const double* __restrict__ sums,
                                const unsigned* __restrict__ prefix,
                                const unsigned* __restrict__ remaining,
                                float* __restrict__ out) {
  if (threadIdx.x == 0 && blockIdx.x == 0) {
    double tot = 0.0;
    for (int r = 0; r < (int)ROWS; ++r) {
      const float thv = __uint_as_float(prefix[r]);
      tot += sums[r] + (double)remaining[r] * (double)thv;
    }
    out[0] = (float)(tot / ((double)ROWS * (double)TOPN));
  }
}

// ---------------------------------------------------------------------------
extern "C" void kernel_launch(void* const* d_in, const int* in_sizes, int n_in,
                              void* d_out, int out_size, void* d_ws, size_t ws_size,
                              hipStream_t stream) {
  (void)in_sizes; (void)n_in; (void)out_size; (void)ws_size;
  const float* x = (const float*)d_in[0];      // net_output  (f32, 16,777,216)
  const float* t = (const float*)d_in[1];      // dummy_target(f32, 16,777,216)
  float* out = (float*)d_out;

  unsigned char* ws = (unsigned char*)d_ws;
  const size_t HIST_BYTES = (size_t)ROWS * HBINS * sizeof(unsigned); // 131072
  unsigned* hist      = (unsigned*)ws;
  unsigned* prefix    = (unsigned*)(ws + HIST_BYTES);        // 8 u32
  unsigned* remaining = (unsigned*)(ws + HIST_BYTES + 32);   // 8 u32
  double*   sums      = (double*)  (ws + HIST_BYTES + 64);   // 8 f64 (8B aligned)

  // Zero hist + prefix + remaining + sums (graph-capturable).
  hipMemsetAsync(ws, 0, HIST_BYTES + 64 + ROWS * sizeof(double), stream);

  const dim3 hgrid(TOTAL_ELEMS / 8192u);       // 2048 blocks, 8192 elems each
  const dim3 hblk(256);

  // Level 0: bits[31:20] (values < 1.0 -> bins <= 1016 of 4096)
  hist_kernel<<<hgrid, hblk, 0, stream>>>(x, t, hist, prefix, 0x00000000u, 20u, 0xFFFu);
  select_kernel<<<ROWS, 256, 0, stream>>>(hist, prefix, remaining, 20u, 0, 4096);
  hipMemsetAsync(hist, 0, HIST_BYTES, stream);

  // Level 1: bits[19:8]
  hist_kernel<<<hgrid, hblk, 0, stream>>>(x, t, hist, prefix, 0xFFF00000u, 8u, 0xFFFu);
  select_kernel<<<ROWS, 256, 0, stream>>>(hist, prefix, remaining, 8u, 1, 4096);
  hipMemsetAsync(hist, 0, HIST_BYTES, stream);

  // Level 2: bits[7:0] -> exact 32-bit threshold pattern
  hist_kernel<<<hgrid, hblk, 0, stream>>>(x, t, hist, prefix, 0xFFFFFF00u, 0u, 0xFFu);
  select_kernel<<<ROWS, 256, 0, stream>>>(hist, prefix, remaining, 0u, 2, 256);

  // Pipelined TDM thresholded sum: 512 single-wave blocks, 64 KB dynamic LDS,
  // 8 chunks (32K elements) per block with double-buffered DMA.
  topk_sum_kernel<<<TOTAL_ELEMS / (CHUNK * CPB), 32,
                    4u * CHUNK * sizeof(float), stream>>>(x, t, prefix, sums);

  finalize_kernel<<<1, 32, 0, stream>>>(sums, prefix, remaining, out);
}